// SimpleRetention_86114094284856
// MI455X (gfx1250) — compile-verified
//
#include <hip/hip_runtime.h>

typedef float v2f __attribute__((ext_vector_type(2)));
typedef float v8f __attribute__((ext_vector_type(8)));

#define WAVES_PER_BLOCK 8
#define THREADS (WAVES_PER_BLOCK * 32)

// wave-private LDS producer->consumer ordering: LDS is in-order per wave on
// CDNA5; fence + wave_barrier stops the compiler from reordering and emits
// the dscnt waits. No s_barrier -> safe inside divergent-trip-count loops.
__device__ __forceinline__ void wave_lds_sync() {
  __builtin_amdgcn_fence(__ATOMIC_ACQ_REL, "workgroup");
  __builtin_amdgcn_wave_barrier();
}

// D = A(16x8) * B(8x16) + C  via two V_WMMA_F32_16X16X4_F32 k-steps
__device__ __forceinline__ v8f wmma_k8(const v2f a0, const v2f a1,
                                       const v2f b0, const v2f b1, v8f c) {
  c = __builtin_amdgcn_wmma_f32_16x16x4_f32(false, a0, false, b0, (short)0, c, false, false);
  c = __builtin_amdgcn_wmma_f32_16x16x4_f32(false, a1, false, b1, (short)0, c, false, false);
  return c;
}

// Store a C-layout 16x16 accumulator into a wave-private LDS tile, lane-major:
// T[lane*8 + v] = M[v + 8*(lane>>4)][lane&15]   -> 2x ds_store_b128 per lane.
__device__ __forceinline__ void store_tile(float* __restrict__ T, int lane, v8f c) {
  float4* dst = (float4*)(T + lane * 8);
  dst[0] = make_float4(c[0], c[1], c[2], c[3]);
  dst[1] = make_float4(c[4], c[5], c[6], c[7]);
}

__global__ __launch_bounds__(THREADS)
void retention_wmma_f32_kernel(const float* __restrict__ X,
                               const float* __restrict__ WQ,
                               const float* __restrict__ WK,
                               const float* __restrict__ WV,
                               const float* __restrict__ Dm,
                               float* __restrict__ Out,
                               int npairs) {
  // --- shared memory -------------------------------------------------------
  __shared__ __attribute__((aligned(16))) float sW[3][4][8][8]; // zero-padded W
  __shared__ __attribute__((aligned(16))) float sD[16][16];     // zero-padded D
  __shared__ __attribute__((aligned(16))) float sX[WAVES_PER_BLOCK][392];
  __shared__ __attribute__((aligned(16))) float sT[WAVES_PER_BLOCK][4][256];
  __shared__ __attribute__((aligned(16))) float sO[WAVES_PER_BLOCK][392];

  const int tid  = threadIdx.x;
  const int lane = tid & 31;
  const int wv   = tid >> 5;
  const int h    = lane >> 4;   // lane half
  const int n16  = lane & 15;   // row (A) / col (B,C,D) index
  const int bb   = n16 >> 3;    // batch-half of this lane's index
  const int n7   = n16 & 7;

  // --- stage W (zero-padded 8x8 per chunk) and D (zero-padded 16x16) -------
  for (int i = tid; i < 3 * 4 * 64; i += THREADS) ((float*)sW)[i] = 0.f;
  for (int i = tid; i < 256; i += THREADS) ((float*)sD)[i] = 0.f;
  __syncthreads();
  for (int i = tid; i < 3 * 4 * 49; i += THREADS) {
    int w = i / 196, rem = i % 196, ch = rem / 49, e = rem % 49;
    const float* src = (w == 0) ? WQ : (w == 1) ? WK : WV;
    sW[w][ch][e / 7][e % 7] = src[ch * 49 + e];
  }
  for (int i = tid; i < 196; i += THREADS) sD[i / 14][i % 14] = Dm[i];
  __syncthreads();

  // --- hoist per-chunk W in B-layout and D in C-layout into registers ------
  // B-layout (mirror of A): vgpr v, half h -> k = v + 2h (+4*ks)
  v2f wqB[4][2], wkB[4][2], wvB[4][2];
  float dCv[4][7];
#pragma unroll
  for (int ch = 0; ch < 4; ++ch) {
#pragma unroll
    for (int ks = 0; ks < 2; ++ks) {
      int k0 = 2 * h + 4 * ks;
      v2f q, k, v;
      q.x = sW[0][ch][k0][n7]; q.y = sW[0][ch][k0 + 1][n7];
      k.x = sW[1][ch][k0][n7]; k.y = sW[1][ch][k0 + 1][n7];
      v.x = sW[2][ch][k0][n7]; v.y = sW[2][ch][k0 + 1][n7];
      wqB[ch][ks] = q; wkB[ch][ks] = k; wvB[ch][ks] = v;
    }
    const int rb = (ch >> 1) * 7, cb = (ch & 1) * 7;
#pragma unroll
    for (int v = 0; v < 7; ++v) dCv[ch][v] = sD[rb + v][cb + n7];
  }

  float* T0 = &sT[wv][0][0];
  float* T1 = &sT[wv][1][0];
  float* T2 = &sT[wv][2][0];
  float* T3 = &sT[wv][3][0];
  float* xs = &sX[wv][0];
  float* os = &sO[wv][0];

  // tile read address bases (float units)
  // P1 (A-style, with per-half k offset):  addr = 8*k + p1base
  // P2 (B-style, with per-half k offset):  addr = k + p2base  (b64-friendly)
  const int p1base = 192 * bb + n7;
  const int p2base = 8 * n16 + 128 * bb;

  const bool hikill = (h == 1);  // lanes whose ks==1,v==1 slot is the k==7 pad

  const float E[7] = {0.2f, 0.04f, 0.008f, 0.0016f,
                      0.00032f, 0.000064f, 0.0000128f};

  const int gw = blockIdx.x * WAVES_PER_BLOCK + wv;
  const int nw = gridDim.x * WAVES_PER_BLOCK;

  for (int p = gw; p < npairs; p += nw) {
    // ---- stage the X pair (392 contiguous floats) into LDS, coalesced ----
    const float4* xg = (const float4*)(X + (size_t)p * 392);
#pragma unroll
    for (int t = 0; t < 4; ++t) {
      int i = lane + 32 * t;
      if (i < 98) ((float4*)xs)[i] = xg[i];
    }
    if (p + nw < npairs && lane * 64 < 1568) {
      __builtin_prefetch((const char*)(X + (size_t)(p + nw) * 392) + lane * 64, 0, 1);
    }
    wave_lds_sync();

    v8f rC = {};  // packed recurrent state, C-layout (off-diag junk unused)

#pragma unroll
    for (int ch = 0; ch < 4; ++ch) {
      const int rb = (ch >> 1) * 7, cb = (ch & 1) * 7;

      // ---- x chunk in A-layout (unconditional loads; select kills k==7,
      //      a mask-multiply would propagate NaN garbage) ------------------
      v2f xA0, xA1;
      {
        int r = (n7 == 7) ? 6 : n7;  // rows 7/15 feed discarded D rows
        const float* xrow = xs + bb * 196 + (rb + r) * 14 + cb;
        int k0 = 2 * h;
        float x0 = xrow[k0];
        float x1 = xrow[k0 + 1];
        float x4 = xrow[k0 + 4];
        float x5 = xrow[k0 + 5];       // safe over-read inside LDS block
        xA0.x = x0; xA0.y = x1;
        xA1.x = x4; xA1.y = hikill ? 0.f : x5;
      }

      // ---- Q, K, V projections (shared W, replicated across halves) -----
      v8f z = {};
      v8f qC = wmma_k8(xA0, xA1, wqB[ch][0], wqB[ch][1], z);
      v8f kC = wmma_k8(xA0, xA1, wkB[ch][0], wkB[ch][1], z);
      v8f vC = wmma_k8(xA0, xA1, wvB[ch][0], wvB[ch][1], z);

      // ---- relayout via wave-private LDS tiles ---------------------------
      store_tile(T0, lane, qC);
      store_tile(T1, lane, kC);
      store_tile(T2, lane, vC);
      store_tile(T3, lane, rC);  // r_prev snapshot for the cross term
      wave_lds_sync();

      v2f qA[2], ktB[2], ktA[2], vB[2], rB[2];
#pragma unroll
      for (int ks = 0; ks < 2; ++ks) {
        const int k0 = 2 * h + 4 * ks;
        const bool kill = hikill && (ks == 1);  // k==7 pad slot
        v2f t;
        t.x = T0[p1base + 8 * k0]; t.y = T0[p1base + 8 * (k0 + 1)];
        if (kill) t.y = 0.f;
        qA[ks] = t;                                   // Q as A
        t.x = T1[p1base + 8 * k0]; t.y = T1[p1base + 8 * (k0 + 1)];
        ktB[ks] = t;                                  // K^T as B
        v2f u = *(const v2f*)&T1[p2base + k0];        // K^T as A
        if (kill) u.y = 0.f;
        ktA[ks] = u;
        vB[ks] = *(const v2f*)&T2[p2base + k0];       // V as B
        rB[ks] = *(const v2f*)&T3[p2base + k0];       // r_prev as B
      }

      // ---- S = (Q K^T) .* D ----------------------------------------------
      v8f sC = wmma_k8(qA[0], qA[1], ktB[0], ktB[1], z);
#pragma unroll
      for (int v = 0; v < 7; ++v) sC[v] *= dCv[ch][v];

      // ---- r += K^T V  (WMMA accumulates straight into the state) -------
      rC = wmma_k8(ktA[0], ktA[1], vB[0], vB[1], rC);

      // ---- cross = (Q r_prev) .* e  (e folds to per-VGPR constants) -----
      v8f crossC = wmma_k8(qA[0], qA[1], rB[0], rB[1], z);
#pragma unroll
      for (int v = 0; v < 7; ++v) crossC[v] *= E[v];

      // ---- out = S V + cross  (cross rides in as the C operand) ---------
      wave_lds_sync();               // qA/ktB reads of T0/T1 complete
      store_tile(T0, lane, sC);
      wave_lds_sync();
      v2f sA[2];
#pragma unroll
      for (int ks = 0; ks < 2; ++ks) {
        const int k0 = 2 * h + 4 * ks;
        v2f t;
        t.x = T0[p1base + 8 * k0]; t.y = T0[p1base + 8 * (k0 + 1)];
        if (hikill && (ks == 1)) t.y = 0.f;
        sA[ks] = t;
      }
      v8f oC = wmma_k8(sA[0], sA[1], vB[0], vB[1], crossC);

      // ---- stage this chunk's 2x(7x7) output quadrants in LDS -----------
      const bool valid = (h == 0) ? (n16 < 7) : (n16 >= 8 && n16 < 15);
      if (valid) {
        const int col = n16 - 8 * h;
        float* orow = os + h * 196 + rb * 14 + cb + col;
#pragma unroll
        for (int v = 0; v < 7; ++v) orow[v * 14] = oC[v];
      }
    }

    // ---- coalesced writeback of the assembled pair (392 floats) ----------
    wave_lds_sync();
    float4* og = (float4*)(Out + (size_t)p * 392);
#pragma unroll
    for (int t = 0; t < 4; ++t) {
      int i = lane + 32 * t;
      if (i < 98) og[i] = ((const float4*)os)[i];
    }
    wave_lds_sync();  // os/xs reads done before next iteration overwrites
  }
}

extern "C" void kernel_launch(void* const* d_in, const int* in_sizes, int n_in,
                              void* d_out, int out_size, void* d_ws, size_t ws_size,
                              hipStream_t stream) {
  const float* X  = (const float*)d_in[0];
  const float* WQ = (const float*)d_in[1];
  const float* WK = (const float*)d_in[2];
  const float* WV = (const float*)d_in[3];
  const float* Dm = (const float*)d_in[4];
  float* Out = (float*)d_out;

  const int npairs = in_sizes[0] / 392;  // B/2 batch-pairs, 2 per WMMA tile
  int blocks = 1024;                     // 8192 waves -> 8 pairs per wave
  int maxBlocks = (npairs + WAVES_PER_BLOCK - 1) / WAVES_PER_BLOCK;
  if (blocks > maxBlocks) blocks = maxBlocks;
  if (blocks < 1) blocks = 1;

  retention_wmma_f32_kernel<<<blocks, THREADS, 0, stream>>>(X, WQ, WK, WV, Dm,
                                                            Out, npairs);
}